// JVAE_63891933495328
// MI455X (gfx1250) — compile-verified
//
#include <hip/hip_runtime.h>
#include <hip/hip_bf16.h>

// Problem constants (match reference)
#define R_STEPS 8192
#define NX 32
#define NM 64
#define LDA 33   // fwd-kernel LDS leading dim (bank-conflict padding)
#define LDB 36   // bwd-kernel LDS leading dim: 144B rows -> 16B aligned for async B128

typedef __attribute__((ext_vector_type(2))) float v2f;
typedef __attribute__((ext_vector_type(8))) float v8f;

// ---- CDNA5 async global->LDS copy (ASYNCcnt-tracked), 16 bytes per issue ----
__device__ __forceinline__ void async_ld16(void* lds, const void* g) {
  asm volatile("global_load_async_to_lds_b128 %0, %1, off"
               :: "v"((unsigned int)(unsigned long long)lds), "v"(g)
               : "memory");
}
__device__ __forceinline__ void wait_async0() {
  asm volatile("s_wait_asynccnt 0" ::: "memory");
}

// 16x16 f32 tile GEMM over K=32 using V_WMMA_F32_16X16X4_F32 (8 K-steps).
// A row-major (M x K, stride lda). If BT: B given row-major as Bt (N x K),
// i.e. computes A @ Bt^T. Else B row-major (K x N, stride ldb).
template<bool BT>
__device__ __forceinline__ v8f mm16x16_k32(const float* A, int lda, int row0,
                                           const float* B, int ldb, int col0,
                                           int lane) {
  const int m  = lane & 15;          // row (A) / col (B) index within tile
  const int kb = (lane >> 4) << 1;   // K sub-pair select (lanes 16-31 -> K+2,K+3)
  v8f acc = {};
#pragma unroll
  for (int k0 = 0; k0 < 32; k0 += 4) {
    v2f a, b;
    a[0] = A[(row0 + m) * lda + k0 + kb];
    a[1] = A[(row0 + m) * lda + k0 + kb + 1];
    if (BT) {
      b[0] = B[(col0 + m) * ldb + k0 + kb];
      b[1] = B[(col0 + m) * ldb + k0 + kb + 1];
    } else {
      b[0] = B[(k0 + kb) * ldb + col0 + m];
      b[1] = B[(k0 + kb + 1) * ldb + col0 + m];
    }
    acc = __builtin_amdgcn_wmma_f32_16x16x4_f32(false, a, false, b,
                                                (short)0, acc, false, false);
  }
  return acc;
}

// --------------------------------------------------------------------------
// Kernel 0: ap = trans_mat @ trans_prec ; apat = ap @ trans_mat^T
// --------------------------------------------------------------------------
__global__ __launch_bounds__(256, 1)
void precompute_kernel(const float* __restrict__ tm, const float* __restrict__ tp,
                       float* __restrict__ ap_g, float* __restrict__ apat_g) {
  __shared__ float sTM[NX * LDA];
  __shared__ float sTP[NX * LDA];
  __shared__ float sAP[NX * LDA];
  const int tid = threadIdx.x;
  for (int e = tid; e < NX * NX; e += 256) {
    int i = e >> 5, j = e & 31;
    sTM[i * LDA + j] = tm[e];
    sTP[i * LDA + j] = tp[e];
  }
  __syncthreads();
  for (int e = tid; e < NX * NX; e += 256) {
    int i = e >> 5, j = e & 31;
    float s = 0.f;
#pragma unroll
    for (int k = 0; k < NX; ++k) s += sTM[i * LDA + k] * sTP[k * LDA + j];
    sAP[i * LDA + j] = s;
    ap_g[e] = s;
  }
  __syncthreads();
  for (int e = tid; e < NX * NX; e += 256) {
    int i = e >> 5, j = e & 31;
    float s = 0.f;
#pragma unroll
    for (int k = 0; k < NX; ++k) s += sAP[i * LDA + k] * sTM[j * LDA + k];
    apat_g[e] = s;
  }
}

// --------------------------------------------------------------------------
// Kernel 1: forward scan (single workgroup, 128 threads = 4 waves).
// hess is double-buffered in LDS via async copies issued one step ahead.
// --------------------------------------------------------------------------
__global__ __launch_bounds__(128, 1)
void fwd_scan_kernel(const float* __restrict__ hess, const float* __restrict__ grads,
                     const float* __restrict__ init_prec, const float* __restrict__ trans_prec,
                     const float* __restrict__ ap_g, const float* __restrict__ apat_g,
                     float* __restrict__ chols, float* __restrict__ offs,
                     float* __restrict__ us) {
  __shared__ float sM[NX * LDA];     // working matrix -> Cholesky factor L
  __shared__ float sPred[NX * LDA];  // pred_prec carry
  __shared__ float sOff[NX * LDA];   // off carry (row-major, off[c][i])
  __shared__ float sAP[NX * LDA];
  __shared__ float sAPAT[NX * LDA];
  __shared__ float sTP[NX * LDA];
  __shared__ __align__(16) float sH[2][NX * NX];  // async-staged hess (contiguous)
  __shared__ float sT[NX];           // RHS for u-solve
  __shared__ float sU[NX];           // u carry
  __shared__ float sInvD[NX];        // 1 / L[i][i]

  const int tid  = threadIdx.x;
  const int wave = tid >> 5;
  const int lane = tid & 31;

  for (int e = tid; e < NX * NX; e += 128) {
    int i = e >> 5, j = e & 31;
    sPred[i * LDA + j] = init_prec[e];
    sAP[i * LDA + j]   = ap_g[e];
    sAPAT[i * LDA + j] = apat_g[e];
    sTP[i * LDA + j]   = trans_prec[e];
    sOff[i * LDA + j]  = 0.f;
  }
  if (tid < NX) sU[tid] = 0.f;
  // Prologue: async-stage hess[0] (256 x 16B chunks over 128 threads)
#pragma unroll
  for (int q = tid; q < 256; q += 128)
    async_ld16(&sH[0][q * 4], hess + q * 4);
  wait_async0();
  __syncthreads();

  for (int r = 0; r < R_STEPS; ++r) {
    const int b = r & 1;
    const float m = (r < R_STEPS - 1) ? 1.f : 0.f;

    // Issue next step's hess into the other buffer (hidden behind this step)
    if (r + 1 < R_STEPS) {
#pragma unroll
      for (int q = tid; q < 256; q += 128)
        async_ld16(&sH[b ^ 1][q * 4], hess + (size_t)(r + 1) * (NX * NX) + q * 4);
    }

    // Phase 1: build M, RHS t (uses off/u from previous step), prefetch r+2
    for (int e = tid; e < NX * NX; e += 128) {
      int i = e >> 5, j = e & 31;
      sM[i * LDA + j] = sPred[i * LDA + j] + sH[b][e] + m * sAPAT[i * LDA + j];
    }
    if (tid < NX) {
      float s = grads[(size_t)r * NX + tid];
#pragma unroll
      for (int j = 0; j < NX; ++j) s -= sU[j] * sOff[tid * LDA + j];
      sT[tid] = s;
    }
    if (wave == 3 && r + 2 < R_STEPS) {   // 32 lanes x 128B = 4KB hess tile -> L2
      __builtin_prefetch(hess + (size_t)(r + 2) * (NX * NX) + lane * 32, 0, 3);
      if (lane == 0) __builtin_prefetch(grads + (size_t)(r + 2) * NX, 0, 3);
    }
    __syncthreads();

    // Phase 2: Cholesky in LDS by wave 0 (intra-wave DS ops are in-order)
    if (wave == 0) {
      const int i = lane;
      for (int k = 0; k < NX; ++k) {
        float d   = sM[k * LDA + k];           // broadcast read
        float l   = sqrtf(d);
        float inv = 1.f / l;
        if (i == k) { sM[k * LDA + k] = l; sInvD[k] = inv; }
        if (i > k)  sM[i * LDA + k] *= inv;
        for (int j = k + 1; j < NX; ++j)       // uniform bounds, predicated
          if (j <= i) sM[i * LDA + j] -= sM[i * LDA + k] * sM[j * LDA + k];
      }
    }
    __syncthreads();

    // Phase 3: triangular solves + stores (waves specialize)
    if (wave == 0) {
      // off: lane c solves L y = -ap[:,c]; off[c][:] = y
      const int c = lane;
      float y[NX];
#pragma unroll
      for (int i = 0; i < NX; ++i) {
        float s = -sAP[i * LDA + c];
#pragma unroll
        for (int j = 0; j < i; ++j) s -= sM[i * LDA + j] * y[j];
        y[i] = s * sInvD[i];
      }
#pragma unroll
      for (int i = 0; i < NX; ++i) {
        sOff[c * LDA + i] = y[i];
        offs[(size_t)r * (NX * NX) + c * NX + i] = y[i];
      }
    } else if (wave == 1) {
      // u: all lanes redundantly solve L z = t (lockstep, L/t broadcast)
      float z[NX];
#pragma unroll
      for (int i = 0; i < NX; ++i) {
        float s = sT[i];
#pragma unroll
        for (int j = 0; j < i; ++j) s -= sM[i * LDA + j] * z[j];
        z[i] = s * sInvD[i];
      }
      sU[lane] = z[lane];
      us[(size_t)r * NX + lane] = z[lane];
    } else if (wave == 2) {
      // store chol (column `lane` of each row); upper-junk never read
#pragma unroll 4
      for (int i = 0; i < NX; ++i)
        chols[(size_t)r * (NX * NX) + i * NX + lane] = sM[i * LDA + lane];
    }
    __syncthreads();

    // Phase 4: pred = TP - off @ off^T -- 4 WMMA tiles, one per wave
    {
      const int row0 = (wave >> 1) * 16;
      const int col0 = (wave & 1) * 16;
      v8f acc = mm16x16_k32<true>(sOff, LDA, row0, sOff, LDA, col0, lane);
#pragma unroll
      for (int v = 0; v < 8; ++v) {
        int rr = row0 + v + ((lane >> 4) * 8);
        int cc = col0 + (lane & 15);
        sPred[rr * LDA + cc] = sTP[rr * LDA + cc] - acc[v];
      }
    }
    wait_async0();     // next step's hess staged
    __syncthreads();
  }
}

// --------------------------------------------------------------------------
// Kernel 2: backward scan (single workgroup, 256 threads = 8 waves).
// chol/off/g are double-buffered in LDS via async B128 copies issued one
// step ahead; stride LDB=36 keeps rows 16B-aligned and banks conflict-free.
// --------------------------------------------------------------------------
__global__ __launch_bounds__(256, 1)
void bwd_scan_kernel(const float* __restrict__ epsx,
                     const float* __restrict__ chols, const float* __restrict__ offs,
                     const float* __restrict__ us, float* __restrict__ out) {
  __shared__ __align__(16) float sChol[2][NX * LDB];
  __shared__ __align__(16) float sOff[2][NX * LDB];
  __shared__ __align__(16) float sG[2][(NM + 1) * LDB]; // row0 = u_r, rows 1..64 = epsx_r
  __shared__ float sTmp[(NM + 1) * LDB];
  __shared__ float sVW[(NM + 1) * LDB];                 // carry (65 x 32)
  __shared__ float sInvD[NX];

  const int tid  = threadIdx.x;
  const int wave = tid >> 5;
  const int lane = tid & 31;

  // Async-stage all operands of step r into buffer bb (16B chunks).
  auto issue_step = [&](int r, int bb) {
    // chol & off: 256 chunks each; thread t -> chunk t
    {
      const int i = tid >> 3, c = (tid & 7) * 4;
      async_ld16(&sChol[bb][i * LDB + c], chols + (size_t)r * (NX * NX) + i * NX + c);
      async_ld16(&sOff[bb][i * LDB + c],  offs  + (size_t)r * (NX * NX) + i * NX + c);
    }
    // g row 0 = us[r]: 8 chunks
    if (tid < 8)
      async_ld16(&sG[bb][tid * 4], us + (size_t)r * NX + tid * 4);
    // g rows 1..64 = epsx[r]: 512 chunks; thread t -> chunks t, t+256
#pragma unroll
    for (int q = tid; q < 512; q += 256) {
      const int j = q >> 3, c = (q & 7) * 4;
      async_ld16(&sG[bb][(1 + j) * LDB + c], epsx + (size_t)r * (NM * NX) + j * NX + c);
    }
  };

  for (int e = tid; e < (NM + 1) * LDB; e += 256) sVW[e] = 0.f;
  issue_step(R_STEPS - 1, (R_STEPS - 1) & 1);
  wait_async0();
  __syncthreads();

  for (int r = R_STEPS - 1; r >= 0; --r) {
    const int b = r & 1;

    // Issue next step's operands (hidden behind this step's compute)
    if (r > 0) issue_step(r - 1, b ^ 1);
    if (r > 1) {  // warm L2 two steps ahead (128B lines)
      if (wave == 4) __builtin_prefetch(chols + (size_t)(r - 2) * (NX * NX) + lane * 32, 0, 3);
      if (wave == 5) __builtin_prefetch(offs  + (size_t)(r - 2) * (NX * NX) + lane * 32, 0, 3);
      if (wave >= 6) __builtin_prefetch(epsx  + (size_t)(r - 2) * (NM * NX) + (tid - 192) * 32, 0, 3);
      if (tid == 0)  __builtin_prefetch(us    + (size_t)(r - 2) * NX, 0, 3);
    }

    // Phase 2: tmp = g - vw @ off. Rows 0..63 via 8 WMMA tiles (wave -> tile).
    {
      const int row0 = (wave >> 1) * 16;
      const int col0 = (wave & 1) * 16;
      v8f acc = mm16x16_k32<false>(sVW, LDB, row0, sOff[b], LDB, col0, lane);
#pragma unroll
      for (int v = 0; v < 8; ++v) {
        int rr = row0 + v + ((lane >> 4) * 8);
        int cc = col0 + (lane & 15);
        sTmp[rr * LDB + cc] = sG[b][rr * LDB + cc] - acc[v];
      }
    }
    if (wave == 0) {  // row 64 (last epsx row) scalar
      float s = sG[b][NM * LDB + lane];
#pragma unroll
      for (int k = 0; k < NX; ++k) s -= sVW[NM * LDB + k] * sOff[b][k * LDB + lane];
      sTmp[NM * LDB + lane] = s;
    }
    if (wave == 1) sInvD[lane] = 1.f / sChol[b][lane * LDB + lane];
    __syncthreads();

    // Phase 3: back-substitution chol^T x = tmp_row, one thread per row (65)
    if (tid <= NM) {
      const int row = tid;
      float x[NX];
#pragma unroll
      for (int i = NX - 1; i >= 0; --i) {
        float s = sTmp[row * LDB + i];
#pragma unroll
        for (int j = i + 1; j < NX; ++j) s -= sChol[b][j * LDB + i] * x[j];
        x[i] = s * sInvD[i];
      }
#pragma unroll
      for (int i = 0; i < NX; ++i) sVW[row * LDB + i] = x[i];
    }
    __syncthreads();

    // Phase 4: out[r][j][c] = vw[0][c] + vw[1+j][c]
    for (int e = tid; e < NM * NX; e += 256) {
      int j = e >> 5, c = e & 31;
      out[(size_t)r * (NM * NX) + e] = sVW[c] + sVW[(1 + j) * LDB + c];
    }
    wait_async0();    // next step's buffers staged
    __syncthreads();
  }
}

// --------------------------------------------------------------------------
// Host-side launch. Inputs (setup_inputs order):
//   0: x_hessian_diags (R,32,32)  1: x_grads (R,1,32)  2: x_trans_mat (32,32)
//   3: x_trans_prec (32,32)       4: x_init_prec (32,32) 5: epsx (R,64,32)
// Output: (R,64,32) float32.
// Workspace (floats): ap[1024] | apat[1024] | chols[R*1024] | offs[R*1024] | us[R*32]
// --------------------------------------------------------------------------
extern "C" void kernel_launch(void* const* d_in, const int* in_sizes, int n_in,
                              void* d_out, int out_size, void* d_ws, size_t ws_size,
                              hipStream_t stream) {
  (void)in_sizes; (void)n_in; (void)out_size; (void)ws_size;
  const float* hess       = (const float*)d_in[0];
  const float* grads      = (const float*)d_in[1];
  const float* trans_mat  = (const float*)d_in[2];
  const float* trans_prec = (const float*)d_in[3];
  const float* init_prec  = (const float*)d_in[4];
  const float* epsx       = (const float*)d_in[5];
  float* out = (float*)d_out;

  float* wsf   = (float*)d_ws;
  float* ap    = wsf;
  float* apat  = wsf + 1024;
  float* chols = wsf + 2048;
  float* offs  = chols + (size_t)R_STEPS * NX * NX;
  float* us    = offs  + (size_t)R_STEPS * NX * NX;

  precompute_kernel<<<1, 256, 0, stream>>>(trans_mat, trans_prec, ap, apat);
  fwd_scan_kernel<<<1, 128, 0, stream>>>(hess, grads, init_prec, trans_prec,
                                         ap, apat, chols, offs, us);
  bwd_scan_kernel<<<1, 256, 0, stream>>>(epsx, chols, offs, us, out);
}